// IMEXETD_72954314490269
// MI455X (gfx1250) — compile-verified
//
#include <hip/hip_runtime.h>
#include <cstdint>

// ---------------- problem constants ----------------
#define DT_C   0.1f
#define IMG_H  1024
#define IMG_W  1024
#define NB     8
#define NTOT   (NB * IMG_H * IMG_W)        // 8388608 elements
#define TILE   16
#define TPAD   18                           // 16 + halo
#define TILES_X (IMG_W / TILE)              // 64
#define TILES_PER_IMG (TILES_X * TILES_X)   // 4096
#define NTILES (NB * TILES_PER_IMG)         // 32768
#define WAVES_PER_BLOCK 8
#define SPMV_BLOCKS (NTILES / WAVES_PER_BLOCK)  // 4096
#define CG_ITERS 200

typedef float v2f __attribute__((ext_vector_type(2)));
typedef float v8f __attribute__((ext_vector_type(8)));
typedef unsigned int v4u __attribute__((ext_vector_type(4)));
typedef int v4i __attribute__((ext_vector_type(4)));
typedef int v8i __attribute__((ext_vector_type(8)));

// ---------------- Tensor Data Mover: DMA a 16x16 f32 tile (row stride 1024)
// from global into LDS with row padding 16 -> 18 dwords (pitch TPAD). --------
__device__ __forceinline__ void tdm_load_tile(const float* gsrc, unsigned lds_byte_addr) {
    uint64_t ga = (uint64_t)(uintptr_t)gsrc;
    // D# group 0: count=1 | lds_addr | global_addr | type=2
    v4u g0 = { 1u,
               lds_byte_addr,
               (unsigned)ga,
               (unsigned)((ga >> 32) & 0x01FFFFFFu) | 0x80000000u };
    // D# group 1:
    //  w0: data_size=2 (4B), pad_enable=1, pad_interval=3 (16 dw), pad_amount=1 (2 dw)
    //  tensor_dim0 = tensor_dim1 = 1024, tile_dim0 = tile_dim1 = 16, stride0 = 1024
    int w0 = (int)((2u << 16) | (1u << 20) | (3u << 22) | (1u << 25));
    int w1 = (int)(((unsigned)IMG_W & 0xFFFFu) << 16);
    int w2 = (int)(((unsigned)IMG_W >> 16) | (((unsigned)IMG_H & 0xFFFFu) << 16));
    int w3 = (int)(((unsigned)IMG_H >> 16) | (16u << 16));
    v8i g1 = { w0, w1, w2, w3, 16, IMG_W, 0, 0 };
    v4i z4 = { 0, 0, 0, 0 };
#if __clang_major__ >= 23
    v8i z8 = { 0, 0, 0, 0, 0, 0, 0, 0 };
    __builtin_amdgcn_tensor_load_to_lds(g0, g1, z4, z4, z8, 0);
#else
    __builtin_amdgcn_tensor_load_to_lds(g0, g1, z4, z4, 0);
#endif
}

// ---------------- fused stencil SpMV (one 16x16 tile per wave) --------------
// MODE 0 (SPMV):  out1 = Ap = p - DT*D*lap(p);  dot += <p, Ap>
// MODE 1 (INIT):  src = u (=b): r = b - A b = DT*D*lap(b);
//                 out1 = r, out2 = p = r, xout = x = u; dot += <r, r>
template <int MODE>
__global__ void __launch_bounds__(256)
spmv_kernel(const float* __restrict__ src, const float* __restrict__ Dco,
            float* __restrict__ out1, float* out2, float* xout,
            float* __restrict__ dotdst)
{
    __shared__ float sm[WAVES_PER_BLOCK][TPAD * TPAD];
    __shared__ float Wm[TILE * TILE];       // tridiag(1,-2,1), symmetric
    __shared__ float red[WAVES_PER_BLOCK];

    const int lane = threadIdx.x & 31;
    const int wv   = threadIdx.x >> 5;
    const int half = lane >> 4;
    const int l16  = lane & 15;

    const int tileId = blockIdx.x * WAVES_PER_BLOCK + wv;
    const int img = tileId / TILES_PER_IMG;
    const int t   = tileId % TILES_PER_IMG;
    const int ty = t / TILES_X, tx = t % TILES_X;
    const int gy0 = ty * TILE, gx0 = tx * TILE;
    const float* imgp = src + (size_t)img * IMG_H * IMG_W;
    float* S = sm[wv];

    // 1) Kick off async DMA of the 16x16 interior into S[1..16][1..16].
    unsigned lds_addr = (unsigned)(uintptr_t)(&S[1 * TPAD + 1]);
    tdm_load_tile(imgp + (size_t)gy0 * IMG_W + gx0, lds_addr);

    // 2) Materialize W in LDS (one element per thread, once per block).
    {
        const int row = threadIdx.x >> 4, col = threadIdx.x & 15;
        const int d = row - col;
        Wm[threadIdx.x] = (d == 0) ? -2.0f : ((d == 1 || d == -1) ? 1.0f : 0.0f);
    }

    // 3) Halo ring (68 cells) with Neumann (edge-clamped) addressing.
    for (int idx = lane; idx < 68; idx += 32) {
        int r, c;
        if (idx < 18)      { r = 0;        c = idx;      }
        else if (idx < 36) { r = 17;       c = idx - 18; }
        else if (idx < 52) { r = idx - 35; c = 0;        }
        else               { r = idx - 51; c = 17;       }
        int gy = min(max(gy0 - 1 + r, 0), IMG_H - 1);
        int gx = min(max(gx0 - 1 + c, 0), IMG_W - 1);
        S[r * TPAD + c] = imgp[(size_t)gy * IMG_W + gx];
    }

    __syncthreads();                        // Wm visible to all waves
    __builtin_amdgcn_s_wait_tensorcnt(0);   // TDM tile resident in LDS
    asm volatile("" ::: "memory");          // fence compiler reordering

    // 4) lap(T) interior = T*W + W*T via chained V_WMMA_F32_16X16X4_F32.
    //    W is symmetric, so the A-form and B-form operand registers of W are
    //    identical per lane: {W[k0][l16], W[k0+1][l16]}.
    v8f accH = { 0.f, 0.f, 0.f, 0.f, 0.f, 0.f, 0.f, 0.f };
    v8f accV = { 0.f, 0.f, 0.f, 0.f, 0.f, 0.f, 0.f, 0.f };
#pragma unroll
    for (int kb = 0; kb < 4; ++kb) {
        const int k0 = 4 * kb + 2 * half;
        v2f wop = { Wm[k0 * TILE + l16], Wm[(k0 + 1) * TILE + l16] };
        v2f aT  = { S[(1 + l16) * TPAD + 1 + k0], S[(1 + l16) * TPAD + 2 + k0] };
        v2f bT  = { S[(1 + k0) * TPAD + 1 + l16], S[(2 + k0) * TPAD + 1 + l16] };
        accH = __builtin_amdgcn_wmma_f32_16x16x4_f32(false, aT, false, wop,
                                                     (short)0, accH, false, false);
        accV = __builtin_amdgcn_wmma_f32_16x16x4_f32(false, wop, false, bT,
                                                     (short)0, accV, false, false);
    }

    // 5) Halo corrections + elementwise Ap / r, write out, accumulate dot.
    //    C layout: VGPR r holds (M = r + 8*half, N = l16).
    const float topv = S[0 * TPAD + 1 + l16];
    const float botv = S[17 * TPAD + 1 + l16];
    const int   sideC   = (l16 == 15) ? 17 : 0;
    const bool  hasSide = (l16 == 0) || (l16 == 15);

    const size_t gbase = ((size_t)img * IMG_H + (gy0 + 8 * half)) * IMG_W + gx0 + l16;
    float dotp = 0.f;
#pragma unroll
    for (int r = 0; r < 8; ++r) {
        const int M = r + 8 * half;
        float lap = accH[r] + accV[r];
        const float side = S[(1 + M) * TPAD + sideC];
        lap += hasSide ? side : 0.f;
        if (r == 0) lap += (half == 0) ? topv : 0.f;
        if (r == 7) lap += (half == 1) ? botv : 0.f;

        const float pc = S[(1 + M) * TPAD + 1 + l16];
        const size_t gidx = gbase + (size_t)r * IMG_W;
        const float dv = Dco[gidx];
        if (MODE == 0) {
            const float ap = pc - DT_C * dv * lap;
            out1[gidx] = ap;
            dotp += pc * ap;
        } else {
            const float rv = DT_C * dv * lap;   // r = b - A b
            out1[gidx] = rv;
            out2[gidx] = rv;                    // p = r
            xout[gidx] = pc;                    // x = b
            dotp += rv * rv;
        }
    }

    // 6) wave32 reduction -> block reduction -> one atomic per block
    for (int off = 16; off > 0; off >>= 1) dotp += __shfl_down(dotp, off, 32);
    if (lane == 0) red[wv] = dotp;
    __syncthreads();
    if (threadIdx.x == 0) {
        float tsum = 0.f;
#pragma unroll
        for (int k = 0; k < WAVES_PER_BLOCK; ++k) tsum += red[k];
        atomicAdd(dotdst, tsum);
    }
}

// ---------------- CG vector updates (float4, L2-resident) -------------------
__global__ void __launch_bounds__(256)
update1_kernel(float* __restrict__ x, float* __restrict__ r,
               const float* __restrict__ p, const float* __restrict__ Ap,
               const float* __restrict__ scal, float* rsnew)
{
    __shared__ float red[WAVES_PER_BLOCK];
    const size_t i = ((size_t)blockIdx.x * blockDim.x + threadIdx.x) * 4;
    const float alpha = scal[0] / (scal[1] + 1e-12f);
    float4 xv = *reinterpret_cast<const float4*>(x + i);
    float4 pv = *reinterpret_cast<const float4*>(p + i);
    float4 rv = *reinterpret_cast<const float4*>(r + i);
    float4 av = *reinterpret_cast<const float4*>(Ap + i);
    xv.x += alpha * pv.x; xv.y += alpha * pv.y; xv.z += alpha * pv.z; xv.w += alpha * pv.w;
    rv.x -= alpha * av.x; rv.y -= alpha * av.y; rv.z -= alpha * av.z; rv.w -= alpha * av.w;
    *reinterpret_cast<float4*>(x + i) = xv;
    *reinterpret_cast<float4*>(r + i) = rv;
    float s = rv.x * rv.x + rv.y * rv.y + rv.z * rv.z + rv.w * rv.w;
    for (int off = 16; off > 0; off >>= 1) s += __shfl_down(s, off, 32);
    const int lane = threadIdx.x & 31, wv = threadIdx.x >> 5;
    if (lane == 0) red[wv] = s;
    __syncthreads();
    if (threadIdx.x == 0) {
        float t = 0.f;
#pragma unroll
        for (int k = 0; k < WAVES_PER_BLOCK; ++k) t += red[k];
        atomicAdd(rsnew, t);
    }
}

__global__ void __launch_bounds__(256)
update2_kernel(float* __restrict__ p, const float* __restrict__ r,
               const float* __restrict__ scal)
{
    const size_t i = ((size_t)blockIdx.x * blockDim.x + threadIdx.x) * 4;
    const float beta = scal[2] / scal[0];
    float4 pv = *reinterpret_cast<const float4*>(p + i);
    float4 rv = *reinterpret_cast<const float4*>(r + i);
    pv.x = rv.x + beta * pv.x; pv.y = rv.y + beta * pv.y;
    pv.z = rv.z + beta * pv.z; pv.w = rv.w + beta * pv.w;
    *reinterpret_cast<float4*>(p + i) = pv;
}

// ---------------- scalar bookkeeping (keeps graph host-sync free) -----------
__global__ void scal_start(float* s) {
    if (threadIdx.x == 0) { s[0] = 0.f; s[1] = 0.f; s[2] = 0.f; }
}
__global__ void scal_pre(float* s) {
    if (threadIdx.x == 0) { s[0] = s[2]; s[1] = 0.f; s[2] = 0.f; }
}

// ---------------- final pointwise ETD update (in place on x = d_out) --------
__global__ void __launch_bounds__(256)
etd_kernel(float* __restrict__ x, const float* __restrict__ kk,
           const float* __restrict__ aCp, const float* __restrict__ Ctp)
{
    const size_t i = ((size_t)blockIdx.x * blockDim.x + threadIdx.x) * 4;
    const float k = kk[0];
    const float a = k - aCp[0] * Ctp[0];
    const float b = k;                         // k / K_CAP, K_CAP = 1
    const float e = expf(fminf(fmaxf(a * DT_C, -60.f), 60.f));
    float4 ut = *reinterpret_cast<const float4*>(x + i);
    float4 o;
    {
        const float num = a * ut.x * e, den = b * ut.x * (e - 1.f) + a;
        o.x = fminf(fmaxf((fabsf(den) > 1e-12f) ? (num / den) : ut.x, 0.f), 1.f);
    }
    {
        const float num = a * ut.y * e, den = b * ut.y * (e - 1.f) + a;
        o.y = fminf(fmaxf((fabsf(den) > 1e-12f) ? (num / den) : ut.y, 0.f), 1.f);
    }
    {
        const float num = a * ut.z * e, den = b * ut.z * (e - 1.f) + a;
        o.z = fminf(fmaxf((fabsf(den) > 1e-12f) ? (num / den) : ut.z, 0.f), 1.f);
    }
    {
        const float num = a * ut.w * e, den = b * ut.w * (e - 1.f) + a;
        o.w = fminf(fmaxf((fabsf(den) > 1e-12f) ? (num / den) : ut.w, 0.f), 1.f);
    }
    *reinterpret_cast<float4*>(x + i) = o;
}

// ---------------- driver ----------------------------------------------------
extern "C" void kernel_launch(void* const* d_in, const int* in_sizes, int n_in,
                              void* d_out, int out_size, void* d_ws, size_t ws_size,
                              hipStream_t stream) {
    (void)in_sizes; (void)n_in; (void)out_size; (void)ws_size;
    const float* u  = (const float*)d_in[0];
    const float* Dc = (const float*)d_in[1];
    const float* kk = (const float*)d_in[2];
    const float* aC = (const float*)d_in[3];
    const float* Ct = (const float*)d_in[4];

    float* x = (float*)d_out;                 // x lives in d_out during CG
    char* ws = (char*)d_ws;
    float* scal = (float*)ws;                 // [rs, pAp, rs_new]
    float* r  = (float*)(ws + 256);
    float* p  = r + (size_t)NTOT;
    float* Ap = p + (size_t)NTOT;

    const dim3 blk(256);
    const int vec_blocks = NTOT / (256 * 4);  // 8192

    scal_start<<<1, 32, 0, stream>>>(scal);
    // x = b; r = p = b - A b = DT*D*lap(b); rs -> scal[2]
    spmv_kernel<1><<<SPMV_BLOCKS, blk, 0, stream>>>(u, Dc, r, p, x, &scal[2]);

    for (int it = 0; it < CG_ITERS; ++it) {
        scal_pre<<<1, 32, 0, stream>>>(scal);                 // rs = rs_new; zero accums
        spmv_kernel<0><<<SPMV_BLOCKS, blk, 0, stream>>>(p, Dc, Ap, nullptr, nullptr, &scal[1]);
        update1_kernel<<<vec_blocks, blk, 0, stream>>>(x, r, p, Ap, scal, &scal[2]);
        update2_kernel<<<vec_blocks, blk, 0, stream>>>(p, r, scal);
    }

    etd_kernel<<<vec_blocks, blk, 0, stream>>>(x, kk, aC, Ct);
}